// NeuralAligner_68418829025519
// MI455X (gfx1250) — compile-verified
//
#include <hip/hip_runtime.h>
#include <hip/hip_bf16.h>
#include <math.h>

typedef float v2f __attribute__((ext_vector_type(2)));
typedef float v8f __attribute__((ext_vector_type(8)));

#define Bn 64
#define Nn 256
#define Dd 512
#define BIGF 1e8f
#define VS 258   // LDS row stride for V: (i*258 + (k-i)) mod 64 banks varies with i -> conflict-free diagonals

// ---- WMMA f32 16x16x4 fragment loaders (ISA 7.12.2 layouts) -------------
// A fragment: rows r0..r0+15, k-cols kk..kk+3 of row-major X (leading dim ld).
// Lane l holds X[r0 + (l&15)][kk + 2*(l>>4) + {0,1}] -> one float2 load.
static __device__ __forceinline__ v2f frag_a(const float* __restrict__ X, int ld, int r0, int kk) {
  const int lane = threadIdx.x & 31;
  const float* p = X + (size_t)(r0 + (lane & 15)) * ld + kk + ((lane >> 4) << 1);
  return *reinterpret_cast<const v2f*>(p);
}
// B fragment from row-major W (K x N): lane l holds W[kk + 2*(l>>4)+{0,1}][c0 + (l&15)]
static __device__ __forceinline__ v2f frag_b_rm(const float* __restrict__ W, int ld, int kk, int c0) {
  const int lane = threadIdx.x & 31;
  const int c = c0 + (lane & 15);
  const int k = kk + ((lane >> 4) << 1);
  v2f b;
  b.x = W[(size_t)k * ld + c];
  b.y = W[(size_t)(k + 1) * ld + c];
  return b;
}

// ---- Kernel 1: four projection GEMMs  C = X(16384x512) @ W(512x512) + bias
// z: 0 -> zx = hx@Wm+bm, 1 -> gx = hx@Wg+bg, 2 -> zy = hy@Wm+bm, 3 -> gy = hy@Wg+bg
__global__ __launch_bounds__(256) void proj_gemm(
    const float* __restrict__ hx, const float* __restrict__ hy,
    const float* __restrict__ Wm, const float* __restrict__ bm,
    const float* __restrict__ Wg, const float* __restrict__ bg,
    float* __restrict__ ws) {
  const int wave = threadIdx.x >> 5, lane = threadIdx.x & 31;
  const int row0 = blockIdx.x * 128 + wave * 16;
  const int col0 = blockIdx.y * 16;
  const int z = blockIdx.z;
  const float* X    = (z < 2) ? hx : hy;
  const float* W    = (z & 1) ? Wg : Wm;
  const float* bias = (z & 1) ? bg : bm;
  float* out = ws + (size_t)z * (size_t)(Bn * Nn) * Dd;

  v8f acc = {0.f, 0.f, 0.f, 0.f, 0.f, 0.f, 0.f, 0.f};
  for (int kk = 0; kk < Dd; kk += 4) {
    v2f a = frag_a(X, Dd, row0, kk);
    v2f b = frag_b_rm(W, Dd, kk, col0);
    acc = __builtin_amdgcn_wmma_f32_16x16x4_f32(false, a, false, b, (short)0, acc, false, false);
  }
  const int n = lane & 15, hi = lane >> 4;
  const float bv = bias[col0 + n];
#pragma unroll
  for (int v = 0; v < 8; ++v) {
    const int m = v + (hi << 3);
    out[(size_t)(row0 + m) * Dd + col0 + n] = acc[v] + bv;
  }
}

// ---- Kernel 2: batched NT GEMM + activation epilogue --------------------
// z==0: theta[b] = softplus(zx[b] @ zy[b]^T);  z==1: A[b] = log_sigmoid(gx[b] @ gy[b]^T)
// NT: B[k][c] = Q[c][k] -> B fragment uses the SAME addressing as an A fragment.
__global__ __launch_bounds__(256) void pair_gemm(
    const float* __restrict__ ws, float* __restrict__ out) {
  const int wave = threadIdx.x >> 5, lane = threadIdx.x & 31;
  const int t  = blockIdx.x * 8 + wave;     // 256 tiles of 16x16 per (b,z)
  const int r0 = (t >> 4) * 16, c0 = (t & 15) * 16;
  const int b = blockIdx.y;
  const int z = blockIdx.z;
  const size_t mat = (size_t)Nn * Dd;
  const float* P = ws + (size_t)(z ? 1 : 0) * (mat * Bn) + (size_t)b * mat;  // zx or gx
  const float* Q = ws + (size_t)(z ? 3 : 2) * (mat * Bn) + (size_t)b * mat;  // zy or gy

  v8f acc = {0.f, 0.f, 0.f, 0.f, 0.f, 0.f, 0.f, 0.f};
  for (int kk = 0; kk < Dd; kk += 4) {
    v2f a  = frag_a(P, Dd, r0, kk);
    v2f bb = frag_a(Q, Dd, c0, kk);   // NT B fragment
    acc = __builtin_amdgcn_wmma_f32_16x16x4_f32(false, a, false, bb, (short)0, acc, false, false);
  }
  const size_t BNM = (size_t)Bn * Nn * Nn;
  float* dst = out + (size_t)(1 + z) * BNM + (size_t)b * Nn * Nn;
  const int n = lane & 15, hi = lane >> 4;
#pragma unroll
  for (int v = 0; v < 8; ++v) {
    const int m = v + (hi << 3);
    const float x = acc[v];
    const float l1p = log1pf(expf(-fabsf(x)));
    const float r = (z == 0) ? (fmaxf(x, 0.f) + l1p)    // softplus
                             : (fminf(x, 0.f) - l1p);   // log_sigmoid
    dst[(size_t)(r0 + m) * Nn + c0 + n] = r;
  }
}

// ---- Kernel 3: smooth Needleman-Wunsch forward + gradient ---------------
// One workgroup per batch element; V[(N+1) x (M+1)] lives in 259 KB of the
// 320 KB WGP LDS. 511 forward + 511 backward antidiagonal steps, workgroup
// barrier between steps. E is written straight into the aln output slice.
__global__ __launch_bounds__(256) void nw_dp(const float* __restrict__ thA,
                                             float* __restrict__ aln) {
  extern __shared__ float Vs[];             // (Nn+1) * VS floats
  const int b = blockIdx.x;
  const int t = threadIdx.x;
  const size_t BNM = (size_t)Bn * Nn * Nn;
  const float* th = thA + BNM + (size_t)b * Nn * Nn;       // theta[b]
  const float* Am = thA + 2 * BNM + (size_t)b * Nn * Nn;   // A[b]
  float* E = aln + (size_t)b * Nn * Nn;                    // E[i][j] -> aln[i-1][j-1]

  for (int idx = t; idx < (Nn + 1) * VS; idx += 256) Vs[idx] = -BIGF;
  __syncthreads();
  if (t == 0) Vs[0] = 0.f;                                 // V[0][0] = 0
  __syncthreads();

  // forward: V[i,j] = theta + lse(A + V[i-1,j], V[i-1,j-1], A + V[i,j-1])
  for (int k = 2; k <= 2 * Nn; ++k) {
    const int i0 = max(1, k - Nn), i1 = min(Nn, k - 1);
    const int i = i0 + t;
    if (i <= i1) {
      const int j = k - i;
      const float aij = Am[(i - 1) * Nn + (j - 1)];
      const float t1 = aij + Vs[(i - 1) * VS + j];
      const float t2 =       Vs[(i - 1) * VS + (j - 1)];
      const float t3 = aij + Vs[i * VS + (j - 1)];
      const float m = fmaxf(fmaxf(t1, t2), t3);
      const float l = m + __logf(__expf(t1 - m) + __expf(t2 - m) + __expf(t3 - m));
      Vs[i * VS + j] = th[(i - 1) * Nn + (j - 1)] + l;
    }
    __syncthreads();
  }

  // backward: E[i,j] = sum over successors of E[succ] * softmax weight.
  // Weights recomputed from V using m(i,j) = V[i,j] - theta[i-1,j-1].
  for (int k = 2 * Nn; k >= 2; --k) {
    const int i0 = max(1, k - Nn), i1 = min(Nn, k - 1);
    const int i = i0 + t;
    if (i <= i1) {
      const int j = k - i;
      float e;
      if (i == Nn && j == Nn) {
        e = 1.f;
      } else {
        const float Vij = Vs[i * VS + j];
        e = 0.f;
        if (i < Nn) {                       // up-successor (i+1, j)
          const float w = __expf(Am[i * Nn + (j - 1)] + th[i * Nn + (j - 1)]
                                 + Vij - Vs[(i + 1) * VS + j]);
          e += E[i * Nn + (j - 1)] * w;
        }
        if (i < Nn && j < Nn) {             // diag-successor (i+1, j+1)
          const float w = __expf(th[i * Nn + j] + Vij - Vs[(i + 1) * VS + (j + 1)]);
          e += E[i * Nn + j] * w;
        }
        if (j < Nn) {                       // left-successor (i, j+1)
          const float w = __expf(Am[(i - 1) * Nn + j] + th[(i - 1) * Nn + j]
                                 + Vij - Vs[i * VS + (j + 1)]);
          e += E[(i - 1) * Nn + j] * w;
        }
      }
      E[(i - 1) * Nn + (j - 1)] = e;
    }
    __syncthreads();
  }
}

extern "C" void kernel_launch(void* const* d_in, const int* in_sizes, int n_in,
                              void* d_out, int out_size, void* d_ws, size_t ws_size,
                              hipStream_t stream) {
  (void)in_sizes; (void)n_in; (void)out_size; (void)ws_size;
  const float* hx = (const float*)d_in[0];
  const float* hy = (const float*)d_in[1];
  const float* Wm = (const float*)d_in[2];
  const float* bm = (const float*)d_in[3];
  const float* Wg = (const float*)d_in[4];
  const float* bg = (const float*)d_in[5];
  float* out = (float*)d_out;
  float* ws  = (float*)d_ws;   // needs 4 * 16384*512 floats = 128 MB

  // 1) four projection GEMMs -> ws slots {zx, gx, zy, gy}
  dim3 g1((Bn * Nn) / 128, Dd / 16, 4);     // 128 x 32 x 4 blocks, 8 waves each
  proj_gemm<<<g1, 256, 0, stream>>>(hx, hy, Wm, bm, Wg, bg, ws);

  // 2) batched NT GEMMs + activations -> theta/A slices of d_out
  dim3 g2(((Nn / 16) * (Nn / 16)) / 8, Bn, 2);  // 32 x 64 x 2
  pair_gemm<<<g2, 256, 0, stream>>>(ws, out);

  // 3) smooth NW forward + gradient; V in LDS (257 x 258 f32 = 259 KB / WGP)
  const size_t smem = (size_t)(Nn + 1) * VS * sizeof(float);
  nw_dp<<<dim3(Bn), 256, smem, stream>>>(out, out);
}